// CrystalXASV1_28097676051004
// MI455X (gfx1250) — compile-verified
//
#include <hip/hip_runtime.h>
#include <hip/hip_bf16.h>

// ---------------------------------------------------------------------------
// CrystalXAS pipeline for MI455X (gfx1250, wave32).
//   feat[N,6] --GIN--> h1[N,128] --GIN--> (fused GEMM+pool) --> pooled[B,512]
//   concat(pooled, prompt@Wp) -> MLP(514->1024->512->100) with batch-norm.
//
// Roofline: the 1.6M-edge x 128-float gather/atomic-scatter (~1.6GB) dominates
// (~70us @ 23.3TB/s); the 13.1-GFLOP GEMM runs on fp32 WMMA
// (V_WMMA_F32_16X16X4_F32) and hides behind it. Pooling is fused into the GEMM
// epilogue (saves the 410MB h2 round trip). GIN eps=0 lets us seed the
// scatter accumulator with h1 itself, making A-panel staging a pure copy ->
// GLOBAL_LOAD_ASYNC_TO_LDS_B128 (ASYNCcnt path).
// ---------------------------------------------------------------------------

typedef float v2f __attribute__((ext_vector_type(2)));
typedef float v8f __attribute__((ext_vector_type(8)));
typedef int   v4i __attribute__((ext_vector_type(4)));

#define FEAT 6
#define H1D 128
#define H2D 512
#define PB 2
#define M0D 1024
#define M1D 512
#define OUTD 100
#define XSTR 520          // 514 cols padded
#define BN_EPS 1e-5f

// ---------------- zero fill ----------------
__global__ void fill_zero_kernel(float* __restrict__ p, long n) {
    long i = (long)blockIdx.x * blockDim.x + threadIdx.x;
    long stride = (long)gridDim.x * blockDim.x;
    for (; i < n; i += stride) p[i] = 0.0f;
}

// ---------------- edge scatter, layer 1 (6 floats/edge) ----------------
__global__ void scatter1_kernel(const float* __restrict__ feat,
                                const int* __restrict__ src,
                                const int* __restrict__ dst,
                                float* __restrict__ neigh1, int E) {
    int e = blockIdx.x * blockDim.x + threadIdx.x;
    if (e >= E) return;
    int s = src[e], d = dst[e];
    const float* fp = feat + (long)s * FEAT;
    float* np = neigh1 + (long)d * FEAT;
#pragma unroll
    for (int j = 0; j < FEAT; ++j) atomicAdd(np + j, fp[j]);
}

// ---- GIN layer 1: h = relu((feat+neigh1)@W0 + b0); written to h1 AND a2buf --
// a2buf is the layer-2 accumulator seeded with h (GIN eps=0: out = h + neigh).
__global__ void gin1_kernel(const float* __restrict__ feat,
                            const float* __restrict__ neigh1,
                            const float* __restrict__ W0,   // [6,128]
                            const float* __restrict__ b0,
                            float* __restrict__ h1,
                            float* __restrict__ a2buf, int nnodes) {
    long idx = (long)blockIdx.x * blockDim.x + threadIdx.x;
    if (idx >= (long)nnodes * H1D) return;
    int node = (int)(idx >> 7);
    int j = (int)(idx & 127);
    float s = b0[j];
#pragma unroll
    for (int k = 0; k < FEAT; ++k)
        s += (feat[(long)node * FEAT + k] + neigh1[(long)node * FEAT + k]) * W0[k * H1D + j];
    s = fmaxf(s, 0.0f);
    h1[idx] = s;
    a2buf[idx] = s;
}

// ---------------- edge scatter, layer 2 (128 floats/edge, wave per edge) ------
__global__ void scatter2_kernel(const float* __restrict__ h1,
                                const int* __restrict__ src,
                                const int* __restrict__ dst,
                                float* __restrict__ a2buf, int E) {
    long tid = (long)blockIdx.x * blockDim.x + threadIdx.x;
    int e = (int)(tid >> 5);
    int lane = (int)(tid & 31);
    if (e >= E) return;
    int s = src[e], d = dst[e];
    const float4 v = *(const float4*)(h1 + (long)s * H1D + lane * 4);
    float* p = a2buf + (long)d * H1D + lane * 4;
    atomicAdd(p + 0, v.x);
    atomicAdd(p + 1, v.y);
    atomicAdd(p + 2, v.z);
    atomicAdd(p + 3, v.w);
}

// ---------------- fused GEMM2 + ReLU + graph pooling (WMMA fp32) --------------
// y = relu(a2buf @ W1 + b1); pooled[gid[r]] += y[r]
// Block: 128 threads (4 waves). Tile: 64 rows x 64 cols, full K=128 in LDS.
// A panel staged via async global->LDS B128 copies; B panel staged transposed.
#define GTM 64
#define GTN 64
#define GKD 128
#define GSTR 132   // padded LDS row stride (dwords): 528B rows -> 16B-aligned,
                   // 8B-aligned b64 reads, conflict-free wmma fragment loads

__global__ __launch_bounds__(128)
void gemm2_pool_kernel(const float* __restrict__ a2,   // [N,128] = h1 + neigh
                       const float* __restrict__ W1,   // [128,512] row-major
                       const float* __restrict__ b1,   // [512]
                       const int* __restrict__ gid,    // [N]
                       float* __restrict__ xbuf,       // [B, XSTR], cols 0..511
                       int nnodes) {
    __shared__ float As[GTM * GSTR];   // [m][k]
    __shared__ float Bs[GTN * GSTR];   // [n][k]  (transposed W1 panel)

    const int m0 = blockIdx.x * GTM;
    const int n0 = blockIdx.y * GTN;
    const int tid = threadIdx.x;

    // Stage A: pure copy a2[m0+m][k] -> As[m][k] via async-to-LDS (ASYNCcnt).
    // Rows past nnodes read in-bounds workspace garbage; masked in epilogue.
    for (int i = tid * 4; i < GTM * GKD; i += 128 * 4) {
        int m = i >> 7, k = i & 127;
        const float* gp = a2 + (long)(m0 + m) * GKD + k;
        float* lp = As + m * GSTR + k;
#if __has_builtin(__builtin_amdgcn_global_load_async_to_lds_b128)
        __builtin_amdgcn_global_load_async_to_lds_b128(
            (__attribute__((address_space(1))) v4i*)(void*)gp,
            (__attribute__((address_space(3))) v4i*)(void*)lp,
            0, 0);
#else
        {
            unsigned ldsoff = (unsigned)(unsigned long long)
                (__attribute__((address_space(3))) float*)lp;
            unsigned long long ga = (unsigned long long)gp;
            asm volatile("global_load_async_to_lds_b128 %0, %1, off"
                         :: "v"(ldsoff), "v"(ga) : "memory");
        }
#endif
    }
    // Stage B transposed: Bs[n][k] = W1[k][n0+n] (global reads coalesced over n)
    for (int i = tid; i < GTN * GKD; i += 128) {
        int k = i >> 6, n = i & 63;
        Bs[n * GSTR + k] = W1[(long)k * H2D + n0 + n];
    }
#if __has_builtin(__builtin_amdgcn_s_wait_asynccnt)
    __builtin_amdgcn_s_wait_asynccnt(0);
#else
    asm volatile("s_wait_asynccnt 0" ::: "memory");
#endif
    __syncthreads();

    const int wave = tid >> 5;
    const int lane = tid & 31;
    const int l15 = lane & 15;
    const int koff = (lane < 16) ? 0 : 2;   // ISA 16x4 f32 A/B layout: K split 0/2

    const float* Arow = As + (wave * 16 + l15) * GSTR;
    v8f acc0 = {}, acc1 = {}, acc2 = {}, acc3 = {};

#pragma unroll 4
    for (int k0 = 0; k0 < GKD; k0 += 4) {
        v2f a = *(const v2f*)(Arow + k0 + koff);                      // A[m][k..k+1]
        v2f bA = *(const v2f*)(Bs + (0 * 16 + l15) * GSTR + k0 + koff);
        v2f bB = *(const v2f*)(Bs + (1 * 16 + l15) * GSTR + k0 + koff);
        v2f bC = *(const v2f*)(Bs + (2 * 16 + l15) * GSTR + k0 + koff);
        v2f bD = *(const v2f*)(Bs + (3 * 16 + l15) * GSTR + k0 + koff);
        acc0 = __builtin_amdgcn_wmma_f32_16x16x4_f32(false, a, false, bA, (short)0, acc0, false, false);
        acc1 = __builtin_amdgcn_wmma_f32_16x16x4_f32(false, a, false, bB, (short)0, acc1, false, false);
        acc2 = __builtin_amdgcn_wmma_f32_16x16x4_f32(false, a, false, bC, (short)0, acc2, false, false);
        acc3 = __builtin_amdgcn_wmma_f32_16x16x4_f32(false, a, false, bD, (short)0, acc3, false, false);
    }

    // Epilogue: bias + relu + atomic pool into xbuf[:, 0:512]
    // C layout: VGPR r -> row r (lanes 0-15) / r+8 (lanes 16-31), col = l15.
    v8f accs[4] = {acc0, acc1, acc2, acc3};
    float bias[4];
#pragma unroll
    for (int t = 0; t < 4; ++t) bias[t] = b1[n0 + t * 16 + l15];
#pragma unroll
    for (int r = 0; r < 8; ++r) {
        int row = m0 + wave * 16 + ((lane < 16) ? r : r + 8);
        if (row >= nnodes) continue;
        int g = gid[row];
#pragma unroll
        for (int t = 0; t < 4; ++t) {
            int col = n0 + t * 16 + l15;
            float val = fmaxf(accs[t][r] + bias[t], 0.0f);
            atomicAdd(xbuf + (long)g * XSTR + col, val);
        }
    }
}

// ---------------- prompt head: x[:,512:514] = prompt@Wp + bp ----------------
__global__ void prompt_kernel(const float* __restrict__ prompt,  // [B,2]
                              const float* __restrict__ Wp,      // [2,2]
                              const float* __restrict__ bp,
                              float* __restrict__ xbuf, int Bsz) {
    int idx = blockIdx.x * blockDim.x + threadIdx.x;
    if (idx >= Bsz * PB) return;
    int b = idx / PB, j = idx % PB;
    float s = bp[j];
#pragma unroll
    for (int k = 0; k < PB; ++k) s += prompt[b * PB + k] * Wp[k * PB + j];
    xbuf[(long)b * XSTR + H2D + j] = s;
}

// ---------------- generic small linear: out = A@W + bias ----------------
__global__ void linear_kernel(const float* __restrict__ A, int Kstride,
                              const float* __restrict__ W,
                              const float* __restrict__ bias,
                              float* __restrict__ out,
                              int rows, int K, int Ncols) {
    long idx = (long)blockIdx.x * blockDim.x + threadIdx.x;
    if (idx >= (long)rows * Ncols) return;
    int r = (int)(idx / Ncols), c = (int)(idx % Ncols);
    float s = bias[c];
    for (int k = 0; k < K; ++k) s += A[(long)r * Kstride + k] * W[(long)k * Ncols + c];
    out[(long)r * Ncols + c] = s;
}

// ---------------- batch-norm (batch stats) + relu, in place ----------------
__global__ __launch_bounds__(128)
void bn_relu_kernel(float* __restrict__ z,
                    const float* __restrict__ gamma,
                    const float* __restrict__ beta, int cols) {
    const int rows = 128;
    int c = blockIdx.x;
    int r = threadIdx.x;
    float v = z[(long)r * cols + c];
    __shared__ float s1[128], s2[128];
    s1[r] = v; s2[r] = v * v;
    __syncthreads();
    for (int off = 64; off > 0; off >>= 1) {
        if (r < off) { s1[r] += s1[r + off]; s2[r] += s2[r + off]; }
        __syncthreads();
    }
    float mean = s1[0] * (1.0f / rows);
    float var = s2[0] * (1.0f / rows) - mean * mean;
    float o = gamma[c] * (v - mean) * rsqrtf(var + BN_EPS) + beta[c];
    z[(long)r * cols + c] = fmaxf(o, 0.0f);
}

// ---------------------------------------------------------------------------
extern "C" void kernel_launch(void* const* d_in, const int* in_sizes, int n_in,
                              void* d_out, int out_size, void* d_ws, size_t ws_size,
                              hipStream_t stream) {
    // setup_inputs() order:
    const float* feat   = (const float*)d_in[0];   // [N,6]
    const float* prompt = (const float*)d_in[1];   // [B,2]
    const int*   src    = (const int*)d_in[2];     // [E]
    const int*   dst    = (const int*)d_in[3];     // [E]
    const int*   gid    = (const int*)d_in[4];     // [N]
    const float* W0  = (const float*)d_in[5];
    const float* b0  = (const float*)d_in[6];
    const float* W1  = (const float*)d_in[7];
    const float* b1  = (const float*)d_in[8];
    const float* Wp  = (const float*)d_in[9];
    const float* bp  = (const float*)d_in[10];
    const float* Wm0 = (const float*)d_in[11];
    const float* bm0 = (const float*)d_in[12];
    const float* g0  = (const float*)d_in[13];
    const float* be0 = (const float*)d_in[14];
    const float* Wm1 = (const float*)d_in[15];
    const float* bm1 = (const float*)d_in[16];
    const float* g1  = (const float*)d_in[17];
    const float* be1 = (const float*)d_in[18];
    const float* Wm2 = (const float*)d_in[19];
    const float* bm2 = (const float*)d_in[20];
    float* out = (float*)d_out;

    const int N = in_sizes[0] / FEAT;
    const int E = in_sizes[2];
    const int B = in_sizes[1] / PB;

    // Workspace layout (floats): [neigh1 | xbuf | a2buf | h1 | z0 | z1]
    // (neigh1 + xbuf are the only zero-init regions -> kept contiguous)
    float* ws = (float*)d_ws;
    float* neigh1 = ws;
    float* xbuf   = neigh1 + (long)N * FEAT;
    float* a2buf  = xbuf   + (long)B * XSTR;
    float* h1     = a2buf  + (long)N * H1D;
    float* z0     = h1     + (long)N * H1D;
    float* z1     = z0     + (long)B * M0D;
    (void)ws_size; (void)n_in; (void)out_size;

    // 1. zero accumulation buffers (contiguous: neigh1, xbuf)
    long nzero = (long)N * FEAT + (long)B * XSTR;
    fill_zero_kernel<<<1024, 256, 0, stream>>>(ws, nzero);

    // 2. GIN layer 1 neighbor sum + linear; output seeds both h1 and a2buf
    scatter1_kernel<<<(E + 255) / 256, 256, 0, stream>>>(feat, src, dst, neigh1, E);
    gin1_kernel<<<(int)(((long)N * H1D + 255) / 256), 256, 0, stream>>>(
        feat, neigh1, W0, b0, h1, a2buf, N);

    // 3. GIN layer 2 neighbor sum (wave per edge, 4 floats per lane)
    long s2threads = (long)E * 32;
    scatter2_kernel<<<(int)((s2threads + 255) / 256), 256, 0, stream>>>(h1, src, dst, a2buf, E);

    // 4. fused GEMM [N,128]x[128,512] + relu + graph pooling (fp32 WMMA)
    dim3 ggrid((N + GTM - 1) / GTM, H2D / GTN);
    gemm2_pool_kernel<<<ggrid, 128, 0, stream>>>(a2buf, W1, b1, gid, xbuf, N);

    // 5. prompt head into x[:,512:514]
    prompt_kernel<<<(B * PB + 255) / 256, 256, 0, stream>>>(prompt, Wp, bp, xbuf, B);

    // 6. MLP head: 514 -> 1024 -> 512 -> 100 with batch-norm + relu
    linear_kernel<<<(int)(((long)B * M0D + 255) / 256), 256, 0, stream>>>(
        xbuf, XSTR, Wm0, bm0, z0, B, H2D + PB, M0D);
    bn_relu_kernel<<<M0D, 128, 0, stream>>>(z0, g0, be0, M0D);

    linear_kernel<<<(int)(((long)B * M1D + 255) / 256), 256, 0, stream>>>(
        z0, M0D, Wm1, bm1, z1, B, M0D, M1D);
    bn_relu_kernel<<<M1D, 128, 0, stream>>>(z1, g1, be1, M1D);

    linear_kernel<<<(int)(((long)B * OUTD + 255) / 256), 256, 0, stream>>>(
        z1, M1D, Wm2, bm2, out, B, M1D, OUTD);
}